// GNN_15307263443376
// MI455X (gfx1250) — compile-verified
//
#include <hip/hip_runtime.h>
#include <hip/hip_bf16.h>
#include <math.h>

typedef __attribute__((ext_vector_type(2))) float v2f;
typedef __attribute__((ext_vector_type(8))) float v8f;

#define HDIM 128

// ---------------- aggregation: agg = h  (self term) ----------------
__global__ void copy_init_kernel(const float* __restrict__ h,
                                 float* __restrict__ agg, int n4)
{
    int i = blockIdx.x * blockDim.x + threadIdx.x;
    if (i < n4) ((float4*)agg)[i] = ((const float4*)h)[i];
}

// ---------------- aggregation: agg[dst] += h[src] over edges -------
// one wave (32 lanes) per edge; each lane moves 4 floats (float4 gather,
// 4 L2-resident f32 atomics). 8 edges per 256-thread block.
__global__ void scatter_add_kernel(const float* __restrict__ h,
                                   const long long* __restrict__ ei,
                                   float* __restrict__ agg, int E)
{
    int e = blockIdx.x * (blockDim.x >> 5) + (threadIdx.x >> 5);
    if (e >= E) return;
    int lane = threadIdx.x & 31;
    long long s = ei[e];
    long long d = ei[(long long)E + e];
    float4 v = ((const float4*)(h + s * HDIM))[lane];
    float* ad = agg + d * HDIM + (long long)lane * 4;
    atomicAdd(ad + 0, v.x);
    atomicAdd(ad + 1, v.y);
    atomicAdd(ad + 2, v.z);
    atomicAdd(ad + 3, v.w);
}

// ---------------- fused GEMM + bias + activation -------------------
// C[nrows x NOUT] = act(A[nrows x 128] @ W[128 x NOUT] + bias)
// ACT: 0 = ReLU, 1 = sigmoid.
// Block = 256 threads = 8 wave32; block computes a 128-row slab.
// Each wave: one 16-row strip, full K=128 A fragments held in VGPRs,
// sweeps NOUT/16 column tiles with v_wmma_f32_16x16x4_f32.
template<int NOUT, int ACT>
__global__ __launch_bounds__(256) void gemm_bias_act_kernel(
    const float* __restrict__ A, const float* __restrict__ W,
    const float* __restrict__ bias, float* __restrict__ C, int nrows)
{
    __shared__ float lds_w[HDIM * NOUT];

    // cooperative stage of W into LDS (float4)
    {
        const float4* w4 = (const float4*)W;
        float4* l4 = (float4*)lds_w;
        const int n4 = (HDIM * NOUT) / 4;
        for (int i = threadIdx.x; i < n4; i += 256) l4[i] = w4[i];
    }
    __syncthreads();

    const int wave = threadIdx.x >> 5;
    const int lane = threadIdx.x & 31;
    const int lo   = lane & 15;   // M (for A) / N (for B,C)
    const int hi   = lane >> 4;   // K-half selector

    const int r0 = blockIdx.x * 128 + wave * 16;
    const int m  = r0 + lo;
    // clamp OOB rows to row 0 (their results are masked at the store)
    const float* arow = A + (long long)(m < nrows ? m : 0) * HDIM;

    // preload this wave's 16 x 128 A strip: 32 k-step fragments
    v2f afrag[32];
#pragma unroll
    for (int kk = 0; kk < 32; ++kk) {
        afrag[kk] = *(const v2f*)(arow + kk * 4 + hi * 2);
    }

#pragma unroll
    for (int n0 = 0; n0 < NOUT; n0 += 16) {
        v8f c = {};
#pragma unroll
        for (int kk = 0; kk < 32; ++kk) {
            const int k = kk * 4 + hi * 2;
            v2f b;
            b.x = lds_w[k * NOUT + n0 + lo];
            b.y = lds_w[(k + 1) * NOUT + n0 + lo];
            c = __builtin_amdgcn_wmma_f32_16x16x4_f32(
                    false, afrag[kk], false, b, (short)0, c, false, false);
        }
        const float bv = bias[n0 + lo];
#pragma unroll
        for (int j = 0; j < 8; ++j) {
            const int row = r0 + j + hi * 8;
            if (row < nrows) {
                float v = c[j] + bv;
                if (ACT == 0) v = fmaxf(v, 0.0f);
                else          v = 1.0f / (1.0f + __expf(-v));
                C[(long long)row * NOUT + n0 + lo] = v;
            }
        }
    }
}

extern "C" void kernel_launch(void* const* d_in, const int* in_sizes, int n_in,
                              void* d_out, int out_size, void* d_ws, size_t ws_size,
                              hipStream_t stream)
{
    const float*      x   = (const float*)d_in[0];
    const long long*  ei  = (const long long*)d_in[1];   // int64 edge_index (2,E)
    const float*      w0a = (const float*)d_in[2];
    const float*      b0a = (const float*)d_in[3];
    const float*      w0b = (const float*)d_in[4];
    const float*      b0b = (const float*)d_in[5];
    const float*      wma = (const float*)d_in[6];       // (4,128,128)
    const float*      bma = (const float*)d_in[7];       // (4,128)
    const float*      wmb = (const float*)d_in[8];
    const float*      bmb = (const float*)d_in[9];
    const float*      wl  = (const float*)d_in[10];      // (128,64)
    const float*      bl  = (const float*)d_in[11];
    float*            out = (float*)d_out;

    const int N = in_sizes[0] / HDIM;
    const int E = in_sizes[1] / 2;

    // scratch: 3 node-feature buffers (each N*128 f32)
    float* agg  = (float*)d_ws;
    float* t    = agg + (size_t)N * HDIM;
    float* hbuf = t   + (size_t)N * HDIM;

    dim3 blk(256);
    const int gInit = (N * (HDIM / 4) + 255) / 256;
    const int gScat = (E + 7) / 8;
    const int gGemm = (N + 127) / 128;

    auto agg_step = [&](const float* src) {
        copy_init_kernel<<<gInit, blk, 0, stream>>>(src, agg, N * (HDIM / 4));
        scatter_add_kernel<<<gScat, blk, 0, stream>>>(src, ei, agg, E);
    };

    // layer 0
    agg_step(x);
    gemm_bias_act_kernel<128, 0><<<gGemm, blk, 0, stream>>>(agg, w0a, b0a, t, N);
    gemm_bias_act_kernel<128, 0><<<gGemm, blk, 0, stream>>>(t, w0b, b0b, hbuf, N);

    // middle layers 1..4
    for (int l = 0; l < 4; ++l) {
        agg_step(hbuf);
        gemm_bias_act_kernel<128, 0><<<gGemm, blk, 0, stream>>>(
            agg, wma + (size_t)l * HDIM * HDIM, bma + (size_t)l * HDIM, t, N);
        gemm_bias_act_kernel<128, 0><<<gGemm, blk, 0, stream>>>(
            t, wmb + (size_t)l * HDIM * HDIM, bmb + (size_t)l * HDIM, hbuf, N);
    }

    // last layer: GEMM 128 -> 64 + sigmoid
    agg_step(hbuf);
    gemm_bias_act_kernel<64, 1><<<gGemm, blk, 0, stream>>>(agg, wl, bl, out, N);
}